// diagonal_24386824306957
// MI455X (gfx1250) — compile-verified
//
#include <hip/hip_runtime.h>

// out[b, j] = in[b, j] * k[j]   (B=8192, N=4096, fp32)
// Memory-bound: ~256 MiB of traffic -> ~11.5 us at 23.3 TB/s. No WMMA:
// 0.125 FLOP/byte makes any matrix path pointless. Optimize the stream:
// b128 vector ops, NT hints on the streams, kernel vector in registers,
// async global->LDS staging (CDNA5 path) for the per-block kernel slice.

typedef float v4f __attribute__((ext_vector_type(4)));
typedef int   v4i __attribute__((vector_size(16)));   // matches builtin's V4i

#if defined(__AMDGCN__) && __has_builtin(__builtin_amdgcn_global_load_async_to_lds_b128)
#define DIAG_ASYNC_LDS 1
#else
#define DIAG_ASYNC_LDS 0
#endif

#define AS1 __attribute__((address_space(1)))
#define AS3 __attribute__((address_space(3)))

__global__ __launch_bounds__(256) void diag_scale_v4_kernel(
    const float* __restrict__ in, const float* __restrict__ kern,
    float* __restrict__ out, int nrows, int nk4) {
  const int c4 = blockIdx.x * 256 + (int)threadIdx.x;  // column index in float4 units
  v4f k = {0.f, 0.f, 0.f, 0.f};

#if DIAG_ASYNC_LDS
  // Stage this block's 4 KiB kernel slice via the CDNA5 async copy path
  // (global_load_async_to_lds_b128, tracked with ASYNCcnt), then pull the
  // per-thread float4 into a register for reuse across all rows.
  __shared__ v4f skv[256];
  if (c4 < nk4) {
    __builtin_amdgcn_global_load_async_to_lds_b128(
        (AS1 v4i*)(kern + 4 * (size_t)c4),
        (AS3 v4i*)(&skv[threadIdx.x]),
        /*imm offset=*/0, /*cpol=*/0);
  }
#if __has_builtin(__builtin_amdgcn_s_wait_asynccnt)
  __builtin_amdgcn_s_wait_asynccnt(0);
#else
  asm volatile("s_wait_asynccnt 0" ::: "memory");
#endif
  __syncthreads();
  if (c4 < nk4) k = skv[threadIdx.x];
#else
  if (c4 < nk4) k = *(const v4f*)(kern + 4 * (size_t)c4);
#endif

  if (c4 >= nk4) return;

  const v4f* __restrict__ in4  = (const v4f*)in;
  v4f* __restrict__       out4 = (v4f*)out;
  const size_t rowstride = (size_t)nk4;
  const size_t step      = (size_t)gridDim.y * rowstride;
  size_t idx = (size_t)blockIdx.y * rowstride + (size_t)c4;

  for (int r = blockIdx.y; r < nrows; r += gridDim.y, idx += step) {
    // Hint the next iteration's cacheline (global_prefetch_b8).
    __builtin_prefetch(in4 + idx + step, 0, 0);
    // Streams are single-use: non-temporal so they don't thrash L2.
    v4f x = __builtin_nontemporal_load(in4 + idx);
    v4f y = x * k;
    __builtin_nontemporal_store(y, out4 + idx);
  }
}

// Fallback for shapes where N % 4 != 0 (not hit by the reference shapes).
__global__ __launch_bounds__(256) void diag_scale_scalar_kernel(
    const float* __restrict__ in, const float* __restrict__ kern,
    float* __restrict__ out, long long total, int N) {
  long long i = (long long)blockIdx.x * blockDim.x + threadIdx.x;
  const long long stride = (long long)gridDim.x * blockDim.x;
  for (; i < total; i += stride) {
    out[i] = in[i] * kern[i % N];
  }
}

extern "C" void kernel_launch(void* const* d_in, const int* in_sizes, int n_in,
                              void* d_out, int out_size, void* d_ws, size_t ws_size,
                              hipStream_t stream) {
  const float* in   = (const float*)d_in[0];
  const float* kern = (const float*)d_in[1];
  float* out        = (float*)d_out;

  const long long total = (long long)in_sizes[0];   // B * N
  const int N           = (n_in > 1) ? in_sizes[1] : 1;

  if (N > 0 && (N & 3) == 0 && total % N == 0) {
    const int nk4         = N >> 2;                 // columns in float4 units
    const long long nrows = total / N;
    dim3 block(256, 1, 1);
    const int gx = (nk4 + 255) / 256;               // 4 blocks across columns
    int gy = (int)(nrows < 2048 ? nrows : 2048);    // 8192 blocks total -> ~4 rows/thread
    if (gy < 1) gy = 1;
    dim3 grid((unsigned)gx, (unsigned)gy, 1);
    diag_scale_v4_kernel<<<grid, block, 0, stream>>>(in, kern, out, (int)nrows, nk4);
  } else {
    long long blocks = (total + 255) / 256;
    if (blocks > 32768) blocks = 32768;
    if (blocks < 1) blocks = 1;
    diag_scale_scalar_kernel<<<(unsigned)blocks, 256, 0, stream>>>(in, kern, out, total, N);
  }
}